// HybridModel_75557064671759
// MI455X (gfx1250) — compile-verified
//
#include <hip/hip_runtime.h>
#include <math.h>

#define DEVINL __device__ __forceinline__

// ============================================================================
// Kernel 1: quanv — one 9-qubit (512 complex amp) state per wave32.
// Amplitude index i in [0,512): lane = i>>4 (bits 8..4), r = i&15 (bits 3..0).
// Wire w acts on bit p = 8 - w.
// ============================================================================

DEVINL void ry_lane(float re[16], float im[16], int lane, int mask, float c, float s) {
  float sg = (lane & mask) ? s : -s;
#pragma unroll
  for (int r = 0; r < 16; ++r) {
    float oR = __shfl_xor(re[r], mask, 32);
    float oI = __shfl_xor(im[r], mask, 32);
    re[r] = c * re[r] + sg * oR;
    im[r] = c * im[r] + sg * oI;
  }
}

DEVINL void rx_lane(float re[16], float im[16], int lane, int mask, float c, float s) {
#pragma unroll
  for (int r = 0; r < 16; ++r) {
    float oR = __shfl_xor(re[r], mask, 32);
    float oI = __shfl_xor(im[r], mask, 32);
    re[r] = c * re[r] + s * oI;
    im[r] = c * im[r] - s * oR;
  }
}

DEVINL void rz_lane(float re[16], float im[16], int lane, int mask, float c, float s) {
  float t = (lane & mask) ? -s : s;  // bit==0 -> e^{-ih}, bit==1 -> e^{+ih}
#pragma unroll
  for (int r = 0; r < 16; ++r) {
    float R = re[r], I = im[r];
    re[r] = c * R + t * I;
    im[r] = c * I - t * R;
  }
}

template <int P>
DEVINL void ry_reg(float re[16], float im[16], float c, float s) {
#pragma unroll
  for (int r0 = 0; r0 < 16; ++r0)
    if (!(r0 & (1 << P))) {
      int r1 = r0 | (1 << P);
      float R0 = re[r0], I0 = im[r0], R1 = re[r1], I1 = im[r1];
      re[r0] = c * R0 - s * R1;  im[r0] = c * I0 - s * I1;
      re[r1] = s * R0 + c * R1;  im[r1] = s * I0 + c * I1;
    }
}

template <int P>
DEVINL void rx_reg(float re[16], float im[16], float c, float s) {
#pragma unroll
  for (int r0 = 0; r0 < 16; ++r0)
    if (!(r0 & (1 << P))) {
      int r1 = r0 | (1 << P);
      float R0 = re[r0], I0 = im[r0], R1 = re[r1], I1 = im[r1];
      re[r0] = c * R0 + s * I1;  im[r0] = c * I0 - s * R1;
      re[r1] = c * R1 + s * I0;  im[r1] = c * I1 - s * R0;
    }
}

template <int P>
DEVINL void rz_reg(float re[16], float im[16], float c, float s) {
#pragma unroll
  for (int r = 0; r < 16; ++r) {
    float t = (r & (1 << P)) ? -s : s;
    float R = re[r], I = im[r];
    re[r] = c * R + t * I;
    im[r] = c * I - t * R;
  }
}

// CNOT: control bit is lane bit, target bit is lane bit
DEVINL void cx_ll(float re[16], float im[16], int lane, int cmask, int tmask) {
  bool c1 = (lane & cmask) != 0;
#pragma unroll
  for (int r = 0; r < 16; ++r) {
    float oR = __shfl_xor(re[r], tmask, 32);
    float oI = __shfl_xor(im[r], tmask, 32);
    re[r] = c1 ? oR : re[r];
    im[r] = c1 ? oI : im[r];
  }
}

// control = lane bit, target = register bit P
template <int P>
DEVINL void cx_lr(float re[16], float im[16], int lane, int cmask) {
  bool c1 = (lane & cmask) != 0;
#pragma unroll
  for (int r0 = 0; r0 < 16; ++r0)
    if (!(r0 & (1 << P))) {
      int r1 = r0 | (1 << P);
      float t;
      t = re[r0]; re[r0] = c1 ? re[r1] : re[r0]; re[r1] = c1 ? t : re[r1];
      t = im[r0]; im[r0] = c1 ? im[r1] : im[r0]; im[r1] = c1 ? t : im[r1];
    }
}

// control = register bit PC, target = register bit PT (pure compile-time swap)
template <int PC, int PT>
DEVINL void cx_rr(float re[16], float im[16]) {
#pragma unroll
  for (int r = 0; r < 16; ++r)
    if ((r & (1 << PC)) && !(r & (1 << PT))) {
      int r2 = r | (1 << PT);
      float t;
      t = re[r]; re[r] = re[r2]; re[r2] = t;
      t = im[r]; im[r] = im[r2]; im[r2] = t;
    }
}

// control = register bit PC, target = lane bit (mask). Both shfl partners share
// the same control value, so every lane swaps its PC==1 registers.
template <int PC>
DEVINL void cx_rl(float re[16], float im[16], int tmask) {
#pragma unroll
  for (int r = 0; r < 16; ++r)
    if (r & (1 << PC)) {
      re[r] = __shfl_xor(re[r], tmask, 32);
      im[r] = __shfl_xor(im[r], tmask, 32);
    }
}

#define KP 1524  // padded feats/W row stride (multiple of 4; rows 16B-aligned)
#define KREAL 1521

__global__ __launch_bounds__(256) void quanv_states_kernel(
    const float* __restrict__ x,    // (1024, 28, 28)
    const float* __restrict__ qp,   // (11,)
    float* __restrict__ feats,      // (1024, KP) padded, valid cols [0,1521)
    int M) {
  const int wave = (int)((blockIdx.x * blockDim.x + threadIdx.x) >> 5);
  const int lane = (int)(threadIdx.x & 31);
  if (wave >= M) return;

  const int b = wave / 169;
  const int patch = wave % 169;
  const int py = patch / 13, px = patch % 13;
  const float* img = x + b * 784;

  // --- data-encoding RY angles (wave-uniform; |h| < 0.5 so native sin/cos ok)
  float ic[9], is_[9];
#pragma unroll
  for (int w = 0; w < 9; ++w) {
    const int i = w / 3, j = w % 3;
    float h = 0.5f * img[(py * 2 + i) * 28 + (px * 2 + j)];
    ic[w] = __cosf(h);
    is_[w] = __sinf(h);
  }
  // --- parameter rotations cos/sin ----------------------------------------
  float qc[11], qs[11];
#pragma unroll
  for (int i = 0; i < 11; ++i) {
    float h = 0.5f * qp[i];
    qc[i] = __cosf(h);
    qs[i] = __sinf(h);
  }

  // --- product-state init: amp(i) = prod_w (bit_{8-w}(i) ? sin : cos) ------
  float F = 1.f;
#pragma unroll
  for (int w = 0; w < 5; ++w)                     // wires 0..4 -> lane bits 4..0
    F *= ((lane >> (4 - w)) & 1) ? is_[w] : ic[w];

  float re[16], im[16];
#pragma unroll
  for (int r = 0; r < 16; ++r) {
    float g = F;
#pragma unroll
    for (int w = 5; w < 9; ++w)                   // wires 5..8 -> reg bits 3..0
      g *= ((r >> (8 - w)) & 1) ? is_[w] : ic[w];
    re[r] = g;
    im[r] = 0.f;
  }

  // --- circuit (wire w -> bit 8-w; bit>=4 -> lane shfl mask 1<<(bit-4)) ----
  ry_lane(re, im, lane, 16, qc[0], qs[0]);   // ry w0  (bit 8)
  rx_lane(re, im, lane, 2,  qc[1], qs[1]);   // rx w3  (bit 5)
  cx_ll  (re, im, lane, 16, 8);              // cnot 0,1 (bits 8->7)
  rz_reg<3>(re, im, qc[2], qs[2]);           // rz w5  (bit 3)
  ry_reg<1>(re, im, qc[3], qs[3]);           // ry w7  (bit 1)
  cx_ll  (re, im, lane, 4, 2);               // cnot 2,3 (bits 6->5)
  rx_lane(re, im, lane, 8,  qc[4], qs[4]);   // rx w1  (bit 7)
  rz_reg<0>(re, im, qc[5], qs[5]);           // rz w8  (bit 0)
  cx_lr<3>(re, im, lane, 1);                 // cnot 4,5 (bit4 -> bit3)
  ry_lane(re, im, lane, 4,  qc[6], qs[6]);   // ry w2  (bit 6)
  rx_reg<2>(re, im, qc[7], qs[7]);           // rx w6  (bit 2)
  cx_rr<2, 1>(re, im);                       // cnot 6,7 (bit2 -> bit1)
  rz_lane(re, im, lane, 1,  qc[8], qs[8]);   // rz w4  (bit 4)
  ry_reg<0>(re, im, qc[9], qs[9]);           // ry w8  (bit 0)
  cx_rl<0>(re, im, 16);                      // cnot 8,0 (bit0 -> bit 8)
  rx_reg<3>(re, im, qc[10], qs[10]);         // rx w5  (bit 3)

  // --- Z expectations: exp[w] = sum_i |amp_i|^2 * (1 - 2*bit_{8-w}(i)) -----
  float T = 0.f, U0 = 0.f, U1 = 0.f, U2 = 0.f, U3 = 0.f;
#pragma unroll
  for (int r = 0; r < 16; ++r) {
    float p = re[r] * re[r] + im[r] * im[r];
    T += p;
    U0 += (r & 1) ? -p : p;
    U1 += (r & 2) ? -p : p;
    U2 += (r & 4) ? -p : p;
    U3 += (r & 8) ? -p : p;
  }
  float vals[9];
  vals[0] = (lane & 16) ? -T : T;   // wire 0 (bit 8 = lane bit 4)
  vals[1] = (lane & 8)  ? -T : T;   // wire 1
  vals[2] = (lane & 4)  ? -T : T;   // wire 2
  vals[3] = (lane & 2)  ? -T : T;   // wire 3
  vals[4] = (lane & 1)  ? -T : T;   // wire 4
  vals[5] = U3;                     // wire 5 (reg bit 3)
  vals[6] = U2;
  vals[7] = U1;
  vals[8] = U0;                     // wire 8 (reg bit 0)

#pragma unroll
  for (int msk = 1; msk <= 16; msk <<= 1) {
#pragma unroll
    for (int w = 0; w < 9; ++w) vals[w] += __shfl_xor(vals[w], msk, 32);
  }

  if (lane == 0) {
#pragma unroll
    for (int w = 0; w < 9; ++w) feats[b * KP + patch * 9 + w] = vals[w];
  }
  // One wave per image zeroes the 3 pad columns so the head kernel can run
  // a guard-free K loop over KP.
  if (patch == 0 && lane < 3) feats[b * KP + KREAL + lane] = 0.f;
}

// ============================================================================
// Prep: zero-padded B image Wp (16, KP): rows 0-1 = W, everything else 0.
// Lets the head loop issue unconditional aligned loads for the B operand.
// ============================================================================
__global__ __launch_bounds__(256) void pad_w_kernel(
    const float* __restrict__ Wt,   // (2, 1521)
    float* __restrict__ wpad) {     // (16, KP)
  const int idx = (int)(blockIdx.x * blockDim.x + threadIdx.x);
  if (idx >= 16 * KP) return;
  const int r = idx / KP;
  const int c = idx % KP;
  float v = 0.f;
  if (r < 2 && c < KREAL) v = Wt[r * KREAL + c];
  wpad[idx] = v;
}

// ============================================================================
// Kernel 2: head — feats(1024x1521) @ W^T(1521x2) + b, log_softmax.
// f32 WMMA 16x16x4: 16 images per wave, N padded to 16, K chunks of 4.
// Loop body = two aligned b64 loads + one v_wmma (fully branch-free).
// ============================================================================

typedef float v2f __attribute__((ext_vector_type(2)));
typedef float v8f __attribute__((ext_vector_type(8)));

__global__ __launch_bounds__(32) void head_kernel(
    const float* __restrict__ feats,  // (1024, KP)
    const float* __restrict__ wpad,   // (16, KP)
    const float* __restrict__ bias,   // (2,)
    float* __restrict__ out) {        // (1024, 2)
  const int lane = (int)(threadIdx.x & 31);
  const int half = lane >> 4;
  const int mn = lane & 15;                  // A: row within tile; B/D: column
  const int row = (int)blockIdx.x * 16 + mn; // image index for A loads

  // A 16x4 layout: lanes 0-15 -> M=lane, VGPR0=K0, VGPR1=K1; lanes 16-31 -> K2,K3.
  // B 4x16 layout: VGPR v, lane half h -> K = v + 2h, N = lane&15.
  const float* arow = feats + (size_t)row * KP + (half << 1);
  const float* brow = wpad + (size_t)mn * KP + (half << 1);

  v8f acc = {};
  for (int kk = 0; kk < KP; kk += 4) {
    v2f a, bm;
    a.x  = arow[kk];
    a.y  = arow[kk + 1];
    bm.x = brow[kk];
    bm.y = brow[kk + 1];
    acc = __builtin_amdgcn_wmma_f32_16x16x4_f32(false, a, false, bm,
                                                (short)0, acc, false, false);
  }

  // D layout: VGPR j, lane half h -> M = j + 8h, N = lane&15. Park in LDS,
  // then 16 lanes finish one image each.
  __shared__ float tile[16 * 16];
#pragma unroll
  for (int j = 0; j < 8; ++j) tile[(j + 8 * half) * 16 + mn] = acc[j];
  __syncthreads();

  if (lane < 16) {
    const int img = (int)blockIdx.x * 16 + lane;
    float l0 = tile[lane * 16 + 0] + bias[0];
    float l1 = tile[lane * 16 + 1] + bias[1];
    float mx = fmaxf(l0, l1);
    float lse = mx + __logf(__expf(l0 - mx) + __expf(l1 - mx));
    out[2 * img + 0] = l0 - lse;
    out[2 * img + 1] = l1 - lse;
  }
}

// ============================================================================
extern "C" void kernel_launch(void* const* d_in, const int* in_sizes, int n_in,
                              void* d_out, int out_size, void* d_ws, size_t ws_size,
                              hipStream_t stream) {
  const float* x  = (const float*)d_in[0];  // (1024,1,28,28)
  const float* qp = (const float*)d_in[1];  // (11,)
  const float* Wt = (const float*)d_in[2];  // (2,1521)
  const float* bs = (const float*)d_in[3];  // (2,)
  float* out = (float*)d_out;               // (1024,2)

  float* feats = (float*)d_ws;              // 1024*KP floats = 6.24 MB
  float* wpad  = feats + 1024 * KP;         // + 16*KP floats  = 97.5 KB
  const int M = 1024 * 169;                 // states; one per wave32

  // 8 waves (= 8 states) per 256-thread block.
  quanv_states_kernel<<<(M + 7) / 8, 256, 0, stream>>>(x, qp, feats, M);
  pad_w_kernel<<<(16 * KP + 255) / 256, 256, 0, stream>>>(Wt, wpad);
  head_kernel<<<1024 / 16, 32, 0, stream>>>(feats, wpad, bs, out);
}